// HiPPO_LegT_1924145349285
// MI455X (gfx1250) — compile-verified
//
#include <hip/hip_runtime.h>

typedef float v2f __attribute__((ext_vector_type(2)));
typedef float v4f __attribute__((ext_vector_type(4)));
typedef float v8f __attribute__((ext_vector_type(8)));

#define NSTATE 256
#define LSEQ   512
#define BATCHD 1024    // batch * D
#define NSEQ   16      // sequences per workgroup
#define SPITCH 260     // padded float pitch of transposed state [seq][row]
#define FCHUNK 256     // timesteps of f staged in LDS at a time

// S_t = A * S_{t-1} + f_t (outer) B, S is 256(state) x 16(seq) per workgroup.
// Wave w (of 16) owns output-state rows [16w, 16w+16): A-slice held in VGPRs
// (k-loop fully unrolled so the fragment array register-allocates), per step:
// 64 V_WMMA_F32_16X16X4_F32 into two interleaved accumulators. State is
// double-buffered in LDS (transposed, padded pitch) with one barrier/step.
__global__ __launch_bounds__(512, 1)
void hippo_legt_scan(const float* __restrict__ inp,   // (1024, 512)
                     const float* __restrict__ Amat,  // (256, 256)
                     const float* __restrict__ Bvec,  // (256)
                     float* __restrict__ out)         // (512, 1024, 256)
{
    __shared__ float fbuf[FCHUNK * NSEQ];        // [t mod 256][seq]
    __shared__ float sbuf[2][NSEQ * SPITCH];     // [seq][row] (padded pitch)

    const int tid  = threadIdx.x;
    const int lane = tid & 31;
    const int wave = tid >> 5;          // 0..15
    const int lo   = lane & 15;         // WMMA N / M index within tile
    const int hi   = lane >> 4;         // lane-half select
    const int s0   = blockIdx.x * NSEQ; // first sequence of this workgroup

    // ---- stage f chunk 0 into LDS, transposed to [t][seq] ----
    {
        const float* src = inp + (size_t)(s0 + wave) * LSEQ;
        #pragma unroll
        for (int i = 0; i < FCHUNK / 32; ++i) {
            int t = lane + 32 * i;                 // coalesced global read
            fbuf[t * NSEQ + wave] = src[t];
        }
    }

    // ---- load this wave's A rows [16w,16w+16) as WMMA A-operand fragments ----
    // A-operand 16x4 f32 layout: lane l holds (M = l%16, K = v + 2*(l/16)).
    v2f a_frag[64];
    {
        const float* arow = Amat + (size_t)(wave * 16 + lo) * NSTATE + 2 * hi;
        #pragma unroll
        for (int k = 0; k < 64; ++k)
            a_frag[k] = *(const v2f*)(arow + 4 * k);   // 8B-aligned, L2-resident
    }
    // B vector rows in D-fragment layout (M = v + 8*hi)
    float brow[8];
    #pragma unroll
    for (int v = 0; v < 8; ++v)
        brow[v] = Bvec[wave * 16 + v + 8 * hi];

    __syncthreads();

    // ---- t = 0: c_0 = f_0 * B ----
    {
        float f = fbuf[lo];
        float* sb = sbuf[0] + lo * SPITCH + wave * 16 + 8 * hi;
        #pragma unroll
        for (int v = 0; v < 8; v += 2) {
            v2f pr; pr.x = f * brow[v]; pr.y = f * brow[v + 1];
            *(v2f*)(sb + v) = pr;                  // 8B-aligned ds_store_b64
        }
    }
    __syncthreads();

    int cur = 0;
    for (int t = 1; t < LSEQ; ++t) {
        if ((t & (FCHUNK - 1)) == 0) {
            // refill f chunk (all prior fbuf reads completed before last barrier)
            const float* src = inp + (size_t)(s0 + wave) * LSEQ + t;
            #pragma unroll
            for (int i = 0; i < FCHUNK / 32; ++i) {
                int tt = lane + 32 * i;
                fbuf[tt * NSEQ + wave] = src[tt];
            }
            __syncthreads();
        }

        // ---- stream state(t-1) to global: coalesced NT float4 stores ----
        {
            const float* sb = sbuf[cur] + wave * SPITCH;
            float* dst = out + ((size_t)(t - 1) * BATCHD + (s0 + wave)) * NSTATE;
            #pragma unroll
            for (int c = 0; c < 2; ++c) {
                v4f val = *(const v4f*)(sb + lane * 4 + 128 * c);
                __builtin_nontemporal_store(val, (v4f*)(dst + lane * 4 + 128 * c));
            }
        }

        // ---- compute state(t) = A * state(t-1) + f_t * B ----
        {
            float f = fbuf[(t & (FCHUNK - 1)) * NSEQ + lo];
            v8f acc0, acc1;
            #pragma unroll
            for (int v = 0; v < 8; ++v) { acc0[v] = f * brow[v]; acc1[v] = 0.0f; }

            // B-operand 4x16 f32 layout: lane l holds (K = v + 2*(l/16), N = l%16)
            const float* sb = sbuf[cur] + lo * SPITCH + 2 * hi;
            #pragma unroll   // FULL unroll: keeps a_frag[] register-resident
            for (int k = 0; k < 64; k += 2) {
                v2f b0 = *(const v2f*)(sb + 4 * k);        // ds_load_b64
                v2f b1 = *(const v2f*)(sb + 4 * (k + 1));
                acc0 = __builtin_amdgcn_wmma_f32_16x16x4_f32(
                         false, a_frag[k],     false, b0, (short)0, acc0, false, false);
                acc1 = __builtin_amdgcn_wmma_f32_16x16x4_f32(
                         false, a_frag[k + 1], false, b1, (short)0, acc1, false, false);
            }
            #pragma unroll
            for (int v = 0; v < 8; ++v) acc0[v] += acc1[v];

            // write new state fragments into the other buffer (transposed)
            float* sd = sbuf[cur ^ 1] + lo * SPITCH + wave * 16 + 8 * hi;
            #pragma unroll
            for (int v = 0; v < 8; v += 2) {
                v2f pr; pr.x = acc0[v]; pr.y = acc0[v + 1];
                *(v2f*)(sd + v) = pr;
            }
        }
        cur ^= 1;
        __syncthreads();   // single barrier per step (double-buffered state)
    }

    // ---- final state (t = 511) ----
    {
        const float* sb = sbuf[cur] + wave * SPITCH;
        float* dst = out + ((size_t)(LSEQ - 1) * BATCHD + (s0 + wave)) * NSTATE;
        #pragma unroll
        for (int c = 0; c < 2; ++c) {
            v4f val = *(const v4f*)(sb + lane * 4 + 128 * c);
            __builtin_nontemporal_store(val, (v4f*)(dst + lane * 4 + 128 * c));
        }
    }
}

extern "C" void kernel_launch(void* const* d_in, const int* in_sizes, int n_in,
                              void* d_out, int out_size, void* d_ws, size_t ws_size,
                              hipStream_t stream) {
    const float* inp  = (const float*)d_in[0];   // (8,128,512) -> (1024,512)
    const float* Amat = (const float*)d_in[1];   // (256,256)
    const float* Bvec = (const float*)d_in[2];   // (256)
    float* out = (float*)d_out;                  // (512,1024,256)

    dim3 grid(BATCHD / NSEQ);   // 64 workgroups
    dim3 block(512);            // 16 waves (wave32)
    hippo_legt_scan<<<grid, block, 0, stream>>>(inp, Amat, Bvec, out);
}